// ARAP_19232863551498
// MI455X (gfx1250) — compile-verified
//
#include <hip/hip_runtime.h>
#include <hip/hip_bf16.h>

// Problem constants from the reference (H=W=160, B=4, D=64).
#define NN   25600
#define BB   4
#define DD   64
#define NPG  8          // nodes per group: one node per wave (256 thr = 8 waves)
#define KCH  (NPG * 6)  // 48 K-rows staged per group
#define LSTR 80         // padded LDS row stride (floats) -> conflict-free frags
#define NSWEEP 10       // Jacobi sweeps

typedef float v2f __attribute__((ext_vector_type(2)));
typedef float v8f __attribute__((ext_vector_type(8)));

// ---------------------------------------------------------------------------
// ws layout:
//   floats [0, BB*64*64)                : Md  (per-batch 64x64 accumulators)
//   ints   at byte off BB*64*64*4       : row_ptr[NN+1]
// ---------------------------------------------------------------------------

__global__ void arap_setup_kernel(const int* __restrict__ e0, int E,
                                  float* __restrict__ Md,
                                  int* __restrict__ rowptr,
                                  float* __restrict__ out) {
    int t = blockIdx.x * blockDim.x + threadIdx.x;
    if (t < BB * DD * DD) Md[t] = 0.0f;
    if (t == 0) out[0] = 0.0f;
    if (t <= NN) {
        // lower_bound over sorted e0 -> CSR row pointer (row_ptr[NN] == E)
        int lo = 0, hi = E;
        while (lo < hi) {
            int mid = (lo + hi) >> 1;
            if (e0[mid] < t) lo = mid + 1; else hi = mid;
        }
        rowptr[t] = lo;
    }
}

// ---------------------------------------------------------------------------
// Fused per-node gather + rank-6 update GEMM via V_WMMA_F32_16X16X4_F32.
// Each wave: computes one node's 6 G-rows (J3[n], BTJ[n]) and 6 H-rows
// (LJ[n], -C^{-1}BTJ[n]), stages to LDS; then owns 2 of the 16 output
// 16x16 tiles of M[b] = G^T H and accumulates them with WMMA over K=48.
// ---------------------------------------------------------------------------
__global__ __launch_bounds__(256)
void arap_fused_kernel(const float* __restrict__ x, const float* __restrict__ J,
                       const int* __restrict__ e1, const int* __restrict__ rowptr,
                       float* __restrict__ Md) {
    __shared__ float Gl[KCH][LSTR];
    __shared__ float Hl[KCH][LSTR];

    const int b    = blockIdx.y;
    const int lane = threadIdx.x & 31;
    const int wave = threadIdx.x >> 5;

    const float* xb = x + (size_t)b * NN * 3;
    const float* Jb = J + (size_t)b * 3 * NN * DD;

    // Tile ownership: wave -> output tile row ti, tile cols {tj0, tj1}.
    const int ti  = (wave & 3) * 16;
    const int tj0 = (wave >> 2) * 32;
    const int tj1 = tj0 + 16;

    v8f acc0 = {0.f, 0.f, 0.f, 0.f, 0.f, 0.f, 0.f, 0.f};
    v8f acc1 = acc0;

    const int ngroups = (NN + NPG - 1) / NPG;
    for (int grp = blockIdx.x; grp < ngroups; grp += gridDim.x) {
        const int n = grp * NPG + wave;

        // Per-lane columns: 2*lane and 2*lane+1 (float2 loads, 256B/row/wave).
        float jn[3][2], nb[3][2], bt[3][2], h0[3][2], h1[3][2];

        if (n < NN) {
            const float xn0 = xb[n * 3 + 0];
            const float xn1 = xb[n * 3 + 1];
            const float xn2 = xb[n * 3 + 2];
#pragma unroll
            for (int r = 0; r < 3; ++r) {
                const float2 jv = ((const float2*)(Jb + ((size_t)n * 3 + r) * DD))[lane];
                jn[r][0] = jv.x; jn[r][1] = jv.y;
                nb[r][0] = 0.f;  nb[r][1] = 0.f;
                bt[r][0] = 0.f;  bt[r][1] = 0.f;
            }
            float c00 = 0.f, c01 = 0.f, c02 = 0.f, c11 = 0.f, c12 = 0.f, c22 = 0.f;

            const int beg = rowptr[n], end = rowptr[n + 1];
            for (int e = beg; e < end; ++e) {
                const int m = e1[e];
                const float v0 = xn0 - xb[m * 3 + 0];
                const float v1 = xn1 - xb[m * 3 + 1];
                const float v2 = xn2 - xb[m * 3 + 2];
                float u[3][2];
#pragma unroll
                for (int r = 0; r < 3; ++r) {
                    const float2 jm = ((const float2*)(Jb + ((size_t)m * 3 + r) * DD))[lane];
                    nb[r][0] += jm.x;            nb[r][1] += jm.y;
                    u[r][0]  = jm.x - jn[r][0];  u[r][1]  = jm.y - jn[r][1];
                }
#pragma unroll
                for (int cc = 0; cc < 2; ++cc) {   // bt += skew(v) @ (Jm - Jn)
                    bt[0][cc] += -v2 * u[1][cc] + v1 * u[2][cc];
                    bt[1][cc] +=  v2 * u[0][cc] - v0 * u[2][cc];
                    bt[2][cc] += -v1 * u[0][cc] + v0 * u[1][cc];
                }
                const float vv = v0 * v0 + v1 * v1 + v2 * v2;
                c00 += vv - v0 * v0; c11 += vv - v1 * v1; c22 += vv - v2 * v2;
                c01 -= v0 * v1; c02 -= v0 * v2; c12 -= v1 * v2;
            }
            const float deg = (float)(end - beg);
#pragma unroll
            for (int r = 0; r < 3; ++r) {          // LJ = 2*(deg*Jn - nb)
                h0[r][0] = 2.f * (deg * jn[r][0] - nb[r][0]);
                h0[r][1] = 2.f * (deg * jn[r][1] - nb[r][1]);
            }
            // symmetric 3x3 inverse (adjugate)
            const float m00 = c11 * c22 - c12 * c12;
            const float m01 = c02 * c12 - c01 * c22;
            const float m02 = c01 * c12 - c02 * c11;
            const float det = c00 * m00 + c01 * m01 + c02 * m02;
            const float rd  = 1.0f / det;
            const float i00 = m00 * rd, i01 = m01 * rd, i02 = m02 * rd;
            const float i11 = (c00 * c22 - c02 * c02) * rd;
            const float i12 = (c01 * c02 - c00 * c12) * rd;
            const float i22 = (c00 * c11 - c01 * c01) * rd;
#pragma unroll
            for (int cc = 0; cc < 2; ++cc) {       // h1 = -(C^{-1} bt)
                h1[0][cc] = -(i00 * bt[0][cc] + i01 * bt[1][cc] + i02 * bt[2][cc]);
                h1[1][cc] = -(i01 * bt[0][cc] + i11 * bt[1][cc] + i12 * bt[2][cc]);
                h1[2][cc] = -(i02 * bt[0][cc] + i12 * bt[1][cc] + i22 * bt[2][cc]);
            }
        } else {
#pragma unroll
            for (int r = 0; r < 3; ++r)
                jn[r][0] = jn[r][1] = bt[r][0] = bt[r][1] =
                h0[r][0] = h0[r][1] = h1[r][0] = h1[r][1] = 0.f;
        }

        __syncthreads();   // previous chunk's fragment reads done
#pragma unroll
        for (int r = 0; r < 3; ++r) {
            Gl[wave * 6 + r][2 * lane]         = jn[r][0];
            Gl[wave * 6 + r][2 * lane + 1]     = jn[r][1];
            Gl[wave * 6 + 3 + r][2 * lane]     = bt[r][0];
            Gl[wave * 6 + 3 + r][2 * lane + 1] = bt[r][1];
            Hl[wave * 6 + r][2 * lane]         = h0[r][0];
            Hl[wave * 6 + r][2 * lane + 1]     = h0[r][1];
            Hl[wave * 6 + 3 + r][2 * lane]     = h1[r][0];
            Hl[wave * 6 + 3 + r][2 * lane + 1] = h1[r][1];
        }
        __syncthreads();

        // WMMA over this K-chunk. A(16x4 f32) layout: lane m=0..15, VGPR v
        // holds K = v + 2*(lane>=16). B(4x16) mirrors it. (ISA 7.12.2)
        const int mrow  = lane & 15;
        const int khalf = lane >> 4;
#pragma unroll
        for (int k = 0; k < KCH; k += 4) {
            v2f a, b0, b1;
            a.x  = Gl[k + 2 * khalf + 0][ti + mrow];
            a.y  = Gl[k + 2 * khalf + 1][ti + mrow];
            b0.x = Hl[k + 2 * khalf + 0][tj0 + mrow];
            b0.y = Hl[k + 2 * khalf + 1][tj0 + mrow];
            b1.x = Hl[k + 2 * khalf + 0][tj1 + mrow];
            b1.y = Hl[k + 2 * khalf + 1][tj1 + mrow];
            acc0 = __builtin_amdgcn_wmma_f32_16x16x4_f32(false, a, false, b0,
                                                         (short)0, acc0, false, false);
            acc1 = __builtin_amdgcn_wmma_f32_16x16x4_f32(false, a, false, b1,
                                                         (short)0, acc1, false, false);
        }
    }

    // Merge partials: C/D layout VGPR r -> M = r + 8*(lane>=16), N = lane&15.
    float* Mb = Md + (size_t)b * DD * DD;
    const int mcol  = lane & 15;
    const int rhalf = (lane >> 4) * 8;
#pragma unroll
    for (int r = 0; r < 8; ++r) {
        atomicAdd(&Mb[(ti + r + rhalf) * DD + tj0 + mcol], acc0[r]);
        atomicAdd(&Mb[(ti + r + rhalf) * DD + tj1 + mcol], acc1[r]);
    }
}

// ---------------------------------------------------------------------------
// 64x64 symmetric eigensolve: parallel round-robin cyclic Jacobi in LDS.
// One workgroup per batch; result: out += sum(sqrt(clip(eig,0))) / B.
// ---------------------------------------------------------------------------
__global__ __launch_bounds__(256)
void arap_eig_kernel(const float* __restrict__ Md, float* __restrict__ out) {
    __shared__ float A[DD][DD + 1];
    __shared__ float Bt[DD][DD + 1];
    __shared__ int   perm[DD];
    __shared__ int   pr[DD];
    __shared__ float cr[DD];
    __shared__ float sr[DD];
    __shared__ float red[256];

    const int b   = blockIdx.x;
    const int tid = threadIdx.x;
    const float* Mb = Md + (size_t)b * DD * DD;

    for (int idx = tid; idx < DD * DD; idx += 256) {
        int i = idx >> 6, j = idx & 63;
        A[i][j] = 0.5f * (Mb[i * DD + j] + Mb[j * DD + i]);  // symmetrize
    }
    for (int i = tid; i < DD; i += 256) perm[i] = i;
    __syncthreads();

    for (int round = 0; round < NSWEEP * (DD - 1); ++round) {
        if (tid < DD / 2) {
            const int p = perm[tid], q = perm[DD - 1 - tid];
            const float apq = A[p][q];
            float c = 1.0f, s = 0.0f;
            if (fabsf(apq) > 1e-30f) {
                const float tau = (A[q][q] - A[p][p]) / (2.0f * apq);
                const float t = (tau >= 0.0f ? 1.0f : -1.0f) /
                                (fabsf(tau) + sqrtf(1.0f + tau * tau));
                c = rsqrtf(1.0f + t * t);
                s = t * c;
            }
            pr[p] = q; pr[q] = p;
            cr[p] = c; cr[q] = c;
            sr[p] = -s; sr[q] = s;
        }
        __syncthreads();
        for (int idx = tid; idx < DD * DD; idx += 256) {        // row pass
            const int r = idx >> 6, col = idx & 63;
            Bt[r][col] = cr[r] * A[r][col] + sr[r] * A[pr[r]][col];
        }
        __syncthreads();
        for (int idx = tid; idx < DD * DD; idx += 256) {        // col pass
            const int r = idx >> 6, col = idx & 63;
            A[r][col] = cr[col] * Bt[r][col] + sr[col] * Bt[r][pr[col]];
        }
        __syncthreads();
        if (tid == 0) {                                          // rotate pairing
            const int last = perm[DD - 1];
            for (int i = DD - 1; i >= 2; --i) perm[i] = perm[i - 1];
            perm[1] = last;
        }
        __syncthreads();
    }

    float local = 0.0f;
    for (int i = tid; i < DD; i += 256) {
        const float lam = A[i][i];
        if (lam > 0.0f) local += sqrtf(lam);
    }
    red[tid] = local;
    __syncthreads();
    for (int st = 128; st > 0; st >>= 1) {
        if (tid < st) red[tid] += red[tid + st];
        __syncthreads();
    }
    if (tid == 0) atomicAdd(out, red[0] * (1.0f / BB));
}

// ---------------------------------------------------------------------------
extern "C" void kernel_launch(void* const* d_in, const int* in_sizes, int n_in,
                              void* d_out, int out_size, void* d_ws, size_t ws_size,
                              hipStream_t stream) {
    const float* x  = (const float*)d_in[0];
    const float* J  = (const float*)d_in[1];
    const int*   e0 = (const int*)d_in[2];
    const int*   e1 = (const int*)d_in[3];
    const int    E  = in_sizes[2];

    float* Md     = (float*)d_ws;
    int*   rowptr = (int*)((char*)d_ws + (size_t)BB * DD * DD * sizeof(float));
    float* out    = (float*)d_out;

    {   // zero accumulators + output, build CSR row pointers
        const int T = NN + 1;  // also covers BB*DD*DD = 16384 zeroing
        arap_setup_kernel<<<(T + 255) / 256, 256, 0, stream>>>(e0, E, Md, rowptr, out);
    }
    {   // fused gather + WMMA rank-update
        dim3 grid(160, BB);
        arap_fused_kernel<<<grid, 256, 0, stream>>>(x, J, e1, rowptr, Md);
    }
    // eigenvalues + trace-sqrt reduction
    arap_eig_kernel<<<BB, 256, 0, stream>>>(Md, out);
}